// NoAuxRouter_14491219657136
// MI455X (gfx1250) — compile-verified
//
#include <hip/hip_runtime.h>
#include <math.h>

#define NEXPERTS 256
#define TOPK     8
#define RSCALE   2.5f

typedef float v4f __attribute__((ext_vector_type(4)));

// ---------------------------------------------------------------------------
// Kernel 0: zero the tokens_per_expert region (graph replays accumulate via
// atomics, so it must be re-zeroed deterministically every launch).
// ---------------------------------------------------------------------------
__global__ __launch_bounds__(NEXPERTS) void zero_hist_kernel(float* __restrict__ hist) {
    hist[threadIdx.x] = 0.0f;
}

// ---------------------------------------------------------------------------
// VALU-only wave32 float max: DPP butterfly within rows + permlanex16 across
// the two 16-lane rows. No LDS traffic (unlike __shfl_xor -> ds_swizzle).
// ---------------------------------------------------------------------------
template <int CTRL>
__device__ __forceinline__ float dpp_maxf(float x) {
    const int y = __builtin_amdgcn_update_dpp(0, __float_as_int(x), CTRL, 0xf, 0xf, true);
    return fmaxf(x, __int_as_float(y));
}

__device__ __forceinline__ float wave32_fmax(float x) {
    x = dpp_maxf<0xB1>(x);   // quad_perm [1,0,3,2]   : lane ^ 1
    x = dpp_maxf<0x4E>(x);   // quad_perm [2,3,0,1]   : lane ^ 2
    x = dpp_maxf<0x141>(x);  // row_half_mirror       : reduce within 8
    x = dpp_maxf<0x140>(x);  // row_mirror            : reduce within 16
    // rows of 16 now uniform; fetch any lane of the other row
    const int y = __builtin_amdgcn_permlanex16(0, __float_as_int(x), -1, -1, true, false);
    return fmaxf(x, __int_as_float(y));
}

// ---------------------------------------------------------------------------
// Router: one wave32 per token; lane L owns experts [8L, 8L+8).
//  - nontemporal b128 loads + write-through echo of logits (streaming, no L2 reuse)
//  - sigmoid + biased choice in registers
//  - 8 extraction rounds: register argmax -> DPP wave max -> ballot/ffs ->
//    v_readlane of winner (idx, score). Zero LDS in the hot loop.
//  - LDS histogram (ds_add_u32) flushed once per block.
// Bias vector staged into LDS via gfx1250 async global->LDS DMA.
// ---------------------------------------------------------------------------
__global__ __launch_bounds__(256, 4) void router_topk_kernel(
    const float* __restrict__ logits,
    const float* __restrict__ bias,
    float* __restrict__ out_logits,
    float* __restrict__ out_w,
    int*   __restrict__ out_idx,
    float* __restrict__ out_hist,
    int T)
{
    __shared__ float        s_bias[NEXPERTS];
    __shared__ unsigned int s_hist[NEXPERTS];

    const int tid  = threadIdx.x;
    const int lane = tid & 31;

    // ---- stage bias[256] (1KB) into LDS: 64 lanes x 16B async DMA ----------
    if (tid < NEXPERTS / 4) {
        unsigned int       lds_addr = (unsigned int)(unsigned long long)(&s_bias[0])
                                      + (unsigned int)tid * 16u;
        unsigned long long gaddr    = (unsigned long long)bias
                                      + (unsigned long long)tid * 16ull;
        asm volatile("global_load_async_to_lds_b128 %0, %1, off"
                     :: "v"(lds_addr), "v"(gaddr) : "memory");
    }
    s_hist[tid] = 0u;
    asm volatile("s_wait_asynccnt 0" ::: "memory");
    __syncthreads();

    const int wave   = (int)((blockIdx.x * blockDim.x + tid) >> 5);
    const int nwaves = (int)((gridDim.x * blockDim.x) >> 5);

    // my 8 bias values (two ds_load_b128)
    const v4f bb0 = ((const v4f*)s_bias)[lane * 2 + 0];
    const v4f bb1 = ((const v4f*)s_bias)[lane * 2 + 1];
    const float bv8[8] = {bb0[0], bb0[1], bb0[2], bb0[3], bb1[0], bb1[1], bb1[2], bb1[3]};

    const float ninf = -__builtin_inff();

    for (int t = wave; t < T; t += nwaves) {
        const size_t rowbase = (size_t)t * NEXPERTS + (size_t)lane * 8;

        // stream-ahead prefetch of this wave's next row (speculative path)
        if (t + nwaves < T)
            __builtin_prefetch(&logits[rowbase + (size_t)nwaves * NEXPERTS], 0, 0);

        const v4f a = __builtin_nontemporal_load((const v4f*)(logits + rowbase));
        const v4f b = __builtin_nontemporal_load((const v4f*)(logits + rowbase) + 1);

        // output #0: logits pass-through (nontemporal b128 stores)
        __builtin_nontemporal_store(a, (v4f*)(out_logits + rowbase));
        __builtin_nontemporal_store(b, (v4f*)(out_logits + rowbase) + 1);

        float sc[8] = {a[0], a[1], a[2], a[3], b[0], b[1], b[2], b[3]};
        float ch[8];
#pragma unroll
        for (int i = 0; i < 8; ++i) {
            const float s = 1.0f / (1.0f + __expf(-sc[i]));  // sigmoid
            sc[i] = s;                                       // raw routing weight
            ch[i] = s + bv8[i];                              // biased selection score
        }

        float sum   = 0.0f;
        float myw   = 0.0f;
        int   myidx = 0;

#pragma unroll
        for (int k = 0; k < TOPK; ++k) {
            // local argmax over my 8 experts (strict > keeps lowest index on tie)
            float bv = ch[0];
            float bs = sc[0];
            int   bi = lane * 8;
#pragma unroll
            for (int i = 1; i < 8; ++i) {
                if (ch[i] > bv) { bv = ch[i]; bs = sc[i]; bi = lane * 8 + i; }
            }
            // VALU-only wave max
            const float wmax = wave32_fmax(bv);
            // lowest lane holding the max == lowest global expert index
            const unsigned long long ball = __ballot(bv == wmax);
            const int srcl = (int)__ffsll(ball) - 1;
            const int   bi_w = __builtin_amdgcn_readlane(bi, srcl);
            const float bs_w = __int_as_float(
                __builtin_amdgcn_readlane(__float_as_int(bs), srcl));

            sum += bs_w;
            if (lane == k) { myw = bs_w; myidx = bi_w; }
            // owner lane retires the winner
#pragma unroll
            for (int i = 0; i < 8; ++i) {
                if (lane * 8 + i == bi_w) ch[i] = ninf;
            }
        }

        const float inv = RSCALE / (sum + 1e-20f);
        if (lane < TOPK) {
            out_w[(size_t)t * TOPK + lane]   = myw * inv;
            out_idx[(size_t)t * TOPK + lane] = myidx;          // raw int32
            atomicAdd(&s_hist[myidx], 1u);                     // ds_add_u32
        }
    }

    __syncthreads();
    // flush per-block histogram (blockDim.x == NEXPERTS)
    const unsigned int h = s_hist[tid];
    if (h) atomicAdd(&out_hist[tid], (float)h);                // exact: integer-valued f32
}

// ---------------------------------------------------------------------------
extern "C" void kernel_launch(void* const* d_in, const int* in_sizes, int n_in,
                              void* d_out, int out_size, void* d_ws, size_t ws_size,
                              hipStream_t stream) {
    (void)n_in; (void)out_size; (void)d_ws; (void)ws_size;

    const float* logits = (const float*)d_in[0];
    const float* bias   = (const float*)d_in[1];
    const int T = in_sizes[0] / NEXPERTS;   // 131072

    float* out_logits = (float*)d_out;
    float* out_w      = out_logits + (size_t)T * NEXPERTS;
    int*   out_idx    = (int*)(out_w + (size_t)T * TOPK);
    float* out_hist   = (float*)(out_idx + (size_t)T * TOPK);

    zero_hist_kernel<<<1, NEXPERTS, 0, stream>>>(out_hist);

    // 256 threads = 8 wave32s per block; grid-stride over tokens.
    int blocks = 2048;
    const int waves_avail = blocks * 8;
    if (waves_avail > T) blocks = (T + 7) / 8;
    router_topk_kernel<<<blocks, 256, 0, stream>>>(
        logits, bias, out_logits, out_w, out_idx, out_hist, T);
}